// CorpusSupportSets_38001870635149
// MI455X (gfx1250) — compile-verified
//
#include <hip/hip_runtime.h>

typedef __attribute__((ext_vector_type(2))) float v2f;
typedef __attribute__((ext_vector_type(4))) float v4f;
typedef __attribute__((ext_vector_type(8))) float v8f;

#define BS   16384
#define KDIM 1000
#define DIM  768

__device__ __forceinline__ float dot4(v4f a, v4f b) {
    return a.x * b.x + a.y * b.y + a.z * b.z + a.w * b.w;
}

// ---------------------------------------------------------------------------
// Kernel 0: pack the [K x 16] parameter matrix B into WMMA "pair" layout.
// Column 0 = k (index extraction), cols 1-2 = ALPHAS[k,:], cols 3-4 = LOGGAMMA[k,:].
// Layout: Bp[(k>>1)*32 + 2*n + (k&1)] so a lane can load {B[kA][n], B[kA+1][n]}
// as one 8-byte float2 (matches V_WMMA_F32_16X16X4_F32 B-matrix VGPR layout).
// ---------------------------------------------------------------------------
__global__ void build_B(const float* __restrict__ ALPHAS,
                        const float* __restrict__ LOGGAMMA,
                        float* __restrict__ Bp) {
    int t = blockIdx.x * blockDim.x + threadIdx.x;
    if (t >= KDIM * 16) return;
    int k = t >> 4;
    int n = t & 15;
    float v = 0.0f;
    if      (n == 0) v = (float)k;
    else if (n == 1) v = ALPHAS[2 * k + 0];
    else if (n == 2) v = ALPHAS[2 * k + 1];
    else if (n == 3) v = LOGGAMMA[2 * k + 0];
    else if (n == 4) v = LOGGAMMA[2 * k + 1];
    Bp[(k >> 1) * 32 + 2 * n + (k & 1)] = v;
}

// ---------------------------------------------------------------------------
// Kernel 1: per 16-sample tile, D = mask_tile(16x1000) @ B(1000x16) using
// V_WMMA_F32_16X16X4_F32, accumulating over 250 K=4 steps.
// D[m][0] = sum_k k*mask[m][k]  -> exact fp32 index (one-hot mask)
// D[m][1..2] = mask @ ALPHAS, D[m][3..4] = mask @ LOGGAMMA (faithful matmul).
// A layout (ISA 7.12.2, 32-bit A 16x4): lanes 0-15 -> M=lane, K={0,1};
// lanes 16-31 -> M=lane-16, K={2,3}. B mirrors with N across lanes.
// ---------------------------------------------------------------------------
__global__ __launch_bounds__(32) void select_wmma(
        const float* __restrict__ mask,
        const float* __restrict__ Bp,
        float* __restrict__ params) {
    const int lane = threadIdx.x & 31;
    const int half = lane >> 4;     // K-half: 0 -> K offset 0, 1 -> K offset 2
    const int m    = lane & 15;     // M (A) / N (B)
    const int tile = blockIdx.x;    // 1024 tiles of 16 samples

    const float* aptr = mask + (size_t)(tile * 16 + m) * KDIM + 2 * half;
    const float* bptr = Bp + half * 32 + 2 * m;

    v8f acc = {0.f, 0.f, 0.f, 0.f, 0.f, 0.f, 0.f, 0.f};

#pragma unroll 2
    for (int k0 = 0; k0 < KDIM; k0 += 4) {
        v2f a = *(const v2f*)(aptr + k0);              // mask[row][kA], mask[row][kA+1]
        v2f b = *(const v2f*)(bptr + (k0 >> 1) * 32);  // B[kA][n],      B[kA+1][n]
        __builtin_prefetch(aptr + k0 + 64, 0, 3);      // global_prefetch_b8, stream mask
        acc = __builtin_amdgcn_wmma_f32_16x16x4_f32(
            /*neg_a=*/false, a, /*neg_b=*/false, b,
            /*c_mod=*/(short)0, acc, /*reuse_a=*/false, /*reuse_b=*/false);
    }

    // D layout: VGPR r, lane -> M = r + (half?8:0), N = lane&15.
    // Persist 8 params per sample: params[s*8 + n].
    if (m < 8) {
        float* p = params + ((size_t)tile * 16 + half * 8) * 8 + m;
#pragma unroll
        for (int r = 0; r < 8; ++r) p[r * 8] = acc[r];
    }
}

// ---------------------------------------------------------------------------
// Kernel 2: one wave32 per sample. Gather dipole SUPPORT_SETS[idx] (L2-hot,
// 6.1 MB total), compute RBF gradient, tangent projection, normalize.
// Bandwidth-bound: 3 KB z in + 6 KB gather + 3 KB out per sample.
// ---------------------------------------------------------------------------
__global__ __launch_bounds__(256) void rbf_project(
        const float* __restrict__ z,
        const float* __restrict__ SUP,
        const float* __restrict__ params,
        float* __restrict__ out) {
    const int lane = threadIdx.x & 31;
    const int wv   = threadIdx.x >> 5;
    const int s    = blockIdx.x * 8 + wv;

    const float* P = params + (size_t)s * 8;
    const float idxf = P[0];
    const float a0 = P[1], a1 = P[2];
    const float g0 = __expf(P[3]), g1 = __expf(P[4]);

    int idx = (int)(idxf + 0.5f);
    idx = idx < 0 ? 0 : (idx > KDIM - 1 ? KDIM - 1 : idx);

    const float* zp  = z   + (size_t)s   * DIM      + lane * 4;
    const float* s0p = SUP + (size_t)idx * (2 * DIM) + lane * 4;
    const float* s1p = s0p + DIM;

    v4f zv[6], d0[6], d1[6];
    float p0 = 0.f, p1 = 0.f;
#pragma unroll
    for (int j = 0; j < 6; ++j) {
        zv[j]  = *(const v4f*)(zp  + j * 128);
        v4f s0 = *(const v4f*)(s0p + j * 128);
        v4f s1 = *(const v4f*)(s1p + j * 128);
        d0[j] = zv[j] - s0;
        d1[j] = zv[j] - s1;
        p0 += dot4(d0[j], d0[j]);
        p1 += dot4(d1[j], d1[j]);
    }
#pragma unroll
    for (int off = 16; off > 0; off >>= 1) {   // wave32 butterfly reduce
        p0 += __shfl_xor(p0, off, 32);
        p1 += __shfl_xor(p1, off, 32);
    }

    const float c0 = -2.0f * a0 * g0 * __expf(-g0 * p0);
    const float c1 = -2.0f * a1 * g1 * __expf(-g1 * p1);

    v4f gv[6];
    float pd = 0.f;
#pragma unroll
    for (int j = 0; j < 6; ++j) {
        gv[j] = c0 * d0[j] + c1 * d1[j];
        pd += dot4(zv[j], gv[j]);
    }
#pragma unroll
    for (int off = 16; off > 0; off >>= 1) pd += __shfl_xor(pd, off, 32);

    float pn = 0.f;
#pragma unroll
    for (int j = 0; j < 6; ++j) {
        gv[j] = gv[j] - pd * zv[j];            // (I - z z^T) grad
        pn += dot4(gv[j], gv[j]);
    }
#pragma unroll
    for (int off = 16; off > 0; off >>= 1) pn += __shfl_xor(pn, off, 32);

    const float inv = rsqrtf(pn);
    float* op = out + (size_t)s * DIM + lane * 4;
#pragma unroll
    for (int j = 0; j < 6; ++j) {
        v4f r = gv[j] * inv;
        *(v4f*)(op + j * 128) = r;             // b128 coalesced store
    }
}

// ---------------------------------------------------------------------------
extern "C" void kernel_launch(void* const* d_in, const int* in_sizes, int n_in,
                              void* d_out, int out_size, void* d_ws, size_t ws_size,
                              hipStream_t stream) {
    const float* mask     = (const float*)d_in[0];  // [BS, K]
    const float* z        = (const float*)d_in[1];  // [BS, DIM]
    const float* SUP      = (const float*)d_in[2];  // [K, 2*DIM]
    const float* ALPHAS   = (const float*)d_in[3];  // [K, 2]
    const float* LOGGAMMA = (const float*)d_in[4];  // [K, 2]
    float* out = (float*)d_out;

    float* Bp     = (float*)d_ws;        // 500 pairs * 32 floats = 64 KB
    float* params = Bp + (KDIM / 2) * 32; // BS * 8 floats = 512 KB

    build_B<<<(KDIM * 16 + 255) / 256, 256, 0, stream>>>(ALPHAS, LOGGAMMA, Bp);
    select_wmma<<<BS / 16, 32, 0, stream>>>(mask, Bp, params);
    rbf_project<<<BS / 8, 256, 0, stream>>>(z, SUP, params, out);
}